// GWTNet_layer_24876450578460
// MI455X (gfx1250) — compile-verified
//
#include <hip/hip_runtime.h>
#include <hip/hip_bf16.h>
#include <math.h>

// ---------------------------------------------------------------------------
// GWTNet-style layer for MI455X (gfx1250), full fp32 via V_WMMA_F32_16X16X4_F32.
// Shapes: B=64, Cin=2, N=512, T=12, D=128, 3 steps, 2 graph-conv steps.
// Intermediate x tensors: [bt, c, n], bt = b*12 + t, row-major 128x512 slices.
// d_out: [b, t, n, d] fp32; each (slice, 16-col) strip is 8KB contiguous.
//
// 4-kernel plan (~3.2 GB HBM traffic; compute ~2us; HBM-bound):
//   k_start_hop2 : start conv (from 3MB raw) -> x ; x@A -> x1 ; x1@A -> x2
//                  (second hop reads A-frags of x1 from LDS; x1 never re-read)
//   k_gc0        : gc + skip0 fused; new x -> P2 in place; skip0 -> out (init)
//   k_hop2       : x@A -> x1 ; x1@A -> x2 (x strip staged in LDS)
//   k_gcF        : gc + skip1 + skip2 + output BN; final x lives in LDS only.
// ---------------------------------------------------------------------------

typedef __attribute__((ext_vector_type(2))) float v2f;
typedef __attribute__((ext_vector_type(8))) float v8f;

#define NBT   768           // B*T
#define DCH   128           // d_model
#define NN    512           // nodes
#define SLICE (DCH * NN)
#define BUFN  (NBT * SLICE) // 50,331,648 floats per tensor
#define EPSBN 1e-5f
#define LSUMP 132           // padded LDS ld for [n][d] sum strip
#define LNXP  20            // padded LDS ld for new-x [c][n] strip
#define LXP   516           // padded LDS ld for 16x512 row strips

__device__ __forceinline__ v8f wmma_f32_k4(v2f a, v2f b, v8f c) {
  return __builtin_amdgcn_wmma_f32_16x16x4_f32(
      false, a, false, b, (short)0, c, false, false);
}

// A-operand: row-major MxK (ld). lane m=l&15; lanes 0-15 K=k0+{0,1}, 16-31 K=k0+{2,3}
__device__ __forceinline__ v2f load_afrag(const float* __restrict__ W, int ld,
                                          int m0, int k0, int mrow, int khalf) {
  const float* p = W + (m0 + mrow) * ld + k0 + 2 * khalf;
  v2f a; a.x = p[0]; a.y = p[1]; return a;
}

// B-operand: row-major KxN (ld). lane n=l&15; VGPR0 K=k0+2*khalf, VGPR1 next row
__device__ __forceinline__ v2f load_bfrag(const float* __restrict__ X, int ld,
                                          int k0, int n0, int ncol, int khalf) {
  const float* p = X + (k0 + 2 * khalf) * ld + n0 + ncol;
  v2f b; b.x = p[0]; b.y = p[ld]; return b;
}

// One graph hop for a 16-row strip: dst(16x512) = src_strip(16x512) @ A(512x512).
// src strip in LDS (ld=LXP); result -> global gdst (ld=NN); STORE_LDS also -> ldst.
template <bool STORE_LDS>
__device__ __forceinline__ void hop_strip(const float* lsrc, const float* __restrict__ A,
                                          float* ldst, float* __restrict__ gdst,
                                          int wave, int mrow, int khalf, int ncol) {
  for (int j = 0; j < 4; ++j) {           // 8 waves x 4 tiles = 32 N-tiles
    int n0 = (wave * 4 + j) * 16;
    v8f acc = {};
#pragma unroll 4
    for (int k0 = 0; k0 < NN; k0 += 4) {
      const float* p = lsrc + mrow * LXP + k0 + 2 * khalf;   // LDS A-frag
      v2f a; a.x = p[0]; a.y = p[1];
      v2f bf = load_bfrag(A, NN, k0, n0, ncol, khalf);       // adj (L2-resident)
      acc = wmma_f32_k4(a, bf, acc);
    }
    int n = n0 + ncol;
#pragma unroll
    for (int v = 0; v < 8; ++v) {
      int m = v + 8 * khalf;
      if (STORE_LDS) ldst[m * LXP + n] = acc[v];
      gdst[m * NN + n] = acc[v];
    }
  }
}

// ---------------------------------------------------------------------------
// Kernel 1: fused start conv + double graph hop.
// Grid: NBT*8 blocks (slice, 16-channel row strip), 256 threads = 8 waves.
// ---------------------------------------------------------------------------
__global__ void k_start_hop2(const float* __restrict__ raw, const float* __restrict__ Ws,
                             const float* __restrict__ bs, const float* __restrict__ A,
                             float* __restrict__ X, float* __restrict__ X1,
                             float* __restrict__ X2) {
  __shared__ float lx[16 * LXP];    // x strip  (33KB)
  __shared__ float lx1[16 * LXP];   // x1 strip (33KB)

  int slice = blockIdx.x >> 3;
  int d0    = (blockIdx.x & 7) * 16;
  int b = slice / 12, t = slice % 12;
  int wave = threadIdx.x >> 5, lane = threadIdx.x & 31;
  int mrow = lane & 15, khalf = lane >> 4, ncol = mrow;

  // start conv for this strip: x[d,n] = W0[d]*raw0[n] + W1[d]*raw1[n] + b[d]
  {
    int dl = threadIdx.x >> 4;      // 0..15 local channel
    int nb = threadIdx.x & 15;
    int d  = d0 + dl;
    float w0 = Ws[d * 2 + 0], w1 = Ws[d * 2 + 1], bb = bs[d];
    const float* r0 = raw + (b * 2 + 0) * NN * 12 + t;
    const float* r1 = raw + (b * 2 + 1) * NN * 12 + t;
    float* gx = X + slice * SLICE + d * NN;
#pragma unroll 4
    for (int n = nb; n < NN; n += 16) {
      float v = w0 * r0[n * 12] + w1 * r1[n * 12] + bb;
      lx[dl * LXP + n] = v;
      gx[n] = v;                    // x -> HBM once (gc0 needs it)
    }
  }
  __syncthreads();

  float* g1 = X1 + slice * SLICE + d0 * NN;
  hop_strip<true>(lx, A, lx1, g1, wave, mrow, khalf, ncol);      // x1 = x @ A
  __syncthreads();

  float* g2 = X2 + slice * SLICE + d0 * NN;
  hop_strip<false>(lx1, A, nullptr, g2, wave, mrow, khalf, ncol); // x2 = x1 @ A
}

// ---------------------------------------------------------------------------
// Kernel 2: double graph hop from materialized x.
// ---------------------------------------------------------------------------
__global__ void k_hop2(const float* __restrict__ X, const float* __restrict__ A,
                       float* __restrict__ X1, float* __restrict__ X2) {
  __shared__ float lx[16 * LXP];
  __shared__ float lx1[16 * LXP];

  int slice = blockIdx.x >> 3;
  int d0    = (blockIdx.x & 7) * 16;
  int wave = threadIdx.x >> 5, lane = threadIdx.x & 31;
  int mrow = lane & 15, khalf = lane >> 4, ncol = mrow;

  // stage x strip: 32KB coalesced float4 loads
  const float* gx = X + slice * SLICE + d0 * NN;
#pragma unroll
  for (int idx = threadIdx.x * 4; idx < 16 * NN; idx += 256 * 4) {
    int row = idx >> 9, col = idx & 511;
    *(float4*)(lx + row * LXP + col) = *(const float4*)(gx + row * NN + col);
  }
  __syncthreads();

  float* g1 = X1 + slice * SLICE + d0 * NN;
  hop_strip<true>(lx, A, lx1, g1, wave, mrow, khalf, ncol);
  __syncthreads();

  float* g2 = X2 + slice * SLICE + d0 * NN;
  hop_strip<false>(lx1, A, nullptr, g2, wave, mrow, khalf, ncol);
}

// ---------------------------------------------------------------------------
// Kernel 3: fused gc step 0 (gc + skip0). new x -> X2 in place; skip0 -> out.
// Grid: NBT*32 blocks (slice, 16-col strip), 256 threads = 8 waves (M-tiles).
// ---------------------------------------------------------------------------
__global__ void k_gc0(const float* __restrict__ X0, const float* __restrict__ X1,
                      float* __restrict__ X2,
                      const float* __restrict__ Wg,  const float* __restrict__ bg,
                      const float* __restrict__ Wsk, const float* __restrict__ bsk,
                      const float* __restrict__ gg, const float* __restrict__ gb,
                      const float* __restrict__ gm, const float* __restrict__ gv,
                      const float* __restrict__ ng, const float* __restrict__ nb,
                      const float* __restrict__ nm, const float* __restrict__ nv,
                      float* __restrict__ out) {
  __shared__ float lsum[16 * LSUMP];

  int slice = blockIdx.x >> 5;
  int cb    = blockIdx.x & 31;
  int wave  = threadIdx.x >> 5;
  int lane  = threadIdx.x & 31;
  int mrow  = lane & 15, khalf = lane >> 4, ncol = mrow;
  int m0 = wave * 16, n0 = cb * 16;

  const float* S0 = X0 + slice * SLICE;
  const float* S1 = X1 + slice * SLICE;
  float*       S2 = X2 + slice * SLICE;

  v8f acc = {}, accs = {};
#pragma unroll 4
  for (int k0 = 0; k0 < DCH; k0 += 4) {        // source 0: x  (+ skip0 conv)
    v2f b  = load_bfrag(S0, NN, k0, n0, ncol, khalf);
    v2f ag = load_afrag(Wg, 3 * DCH, m0, k0, mrow, khalf);
    v2f as = load_afrag(Wsk, DCH, m0, k0, mrow, khalf);
    acc  = wmma_f32_k4(ag, b, acc);
    accs = wmma_f32_k4(as, b, accs);
  }
#pragma unroll 4
  for (int k0 = 0; k0 < DCH; k0 += 4) {        // source 1: A x
    v2f a = load_afrag(Wg, 3 * DCH, m0, DCH + k0, mrow, khalf);
    v2f b = load_bfrag(S1, NN, k0, n0, ncol, khalf);
    acc = wmma_f32_k4(a, b, acc);
  }
#pragma unroll 4
  for (int k0 = 0; k0 < DCH; k0 += 4) {        // source 2: A^2 x
    v2f a = load_afrag(Wg, 3 * DCH, m0, 2 * DCH + k0, mrow, khalf);
    v2f b = load_bfrag(S2, NN, k0, n0, ncol, khalf);
    acc = wmma_f32_k4(a, b, acc);
  }

  int n = n0 + ncol;
  float nx[8];
#pragma unroll
  for (int v = 0; v < 8; ++v) {
    int m = m0 + v + 8 * khalf;
    float gi = gg[m] * rsqrtf(gv[m] + EPSBN);
    float bi = ng[m] * rsqrtf(nv[m] + EPSBN);
    float h = acc[v] + bg[m];
    h = h * gi + (gb[m] - gm[m] * gi);       // gc BN
    h = h + S0[m * NN + n];                  // residual
    nx[v] = h * bi + (nb[m] - nm[m] * bi);   // BN -> new x
    lsum[ncol * LSUMP + m] = accs[v] + bsk[m];
  }

  __syncthreads();  // all S2 reads consumed; lsum staged

#pragma unroll
  for (int v = 0; v < 8; ++v) {              // new x -> X2 in place
    int m = m0 + v + 8 * khalf;
    S2[m * NN + n] = nx[v];
  }

  float* ob = out + (slice * NN + n0) * DCH; // 8KB contiguous strip, init store
#pragma unroll
  for (int j = threadIdx.x * 4; j < 16 * DCH; j += 256 * 4) {
    int nn = j >> 7, dd = j & 127;
    *(float4*)(ob + j) = *(float4*)(lsum + nn * LSUMP + dd);
  }
}

// ---------------------------------------------------------------------------
// Kernel 4: fused gc step 1 + skip1 + skip2 + output BN.
// Final x is staged in LDS only (never written to HBM); d_out finalized.
// ---------------------------------------------------------------------------
__global__ void k_gcF(const float* __restrict__ X0, const float* __restrict__ X1,
                      const float* __restrict__ X2,
                      const float* __restrict__ Wg,   const float* __restrict__ bg,
                      const float* __restrict__ Wsk1, const float* __restrict__ bsk1,
                      const float* __restrict__ Wsk2, const float* __restrict__ bsk2,
                      const float* __restrict__ gg, const float* __restrict__ gb,
                      const float* __restrict__ gm, const float* __restrict__ gv,
                      const float* __restrict__ ng, const float* __restrict__ nb,
                      const float* __restrict__ nm, const float* __restrict__ nv,
                      const float* __restrict__ og, const float* __restrict__ obt,
                      const float* __restrict__ om, const float* __restrict__ ov,
                      float* __restrict__ out) {
  __shared__ float lnx[DCH * LNXP];
  __shared__ float lsum[16 * LSUMP];

  int slice = blockIdx.x >> 5;
  int cb    = blockIdx.x & 31;
  int wave  = threadIdx.x >> 5;
  int lane  = threadIdx.x & 31;
  int mrow  = lane & 15, khalf = lane >> 4, ncol = mrow;
  int m0 = wave * 16, n0 = cb * 16;

  const float* S0 = X0 + slice * SLICE;
  const float* S1 = X1 + slice * SLICE;
  const float* S2 = X2 + slice * SLICE;

  v8f acc = {}, accs = {};
#pragma unroll 4
  for (int k0 = 0; k0 < DCH; k0 += 4) {        // source 0: x  (+ skip1 conv)
    v2f b  = load_bfrag(S0, NN, k0, n0, ncol, khalf);
    v2f ag = load_afrag(Wg, 3 * DCH, m0, k0, mrow, khalf);
    v2f as = load_afrag(Wsk1, DCH, m0, k0, mrow, khalf);
    acc  = wmma_f32_k4(ag, b, acc);
    accs = wmma_f32_k4(as, b, accs);
  }
#pragma unroll 4
  for (int k0 = 0; k0 < DCH; k0 += 4) {        // source 1: A x
    v2f a = load_afrag(Wg, 3 * DCH, m0, DCH + k0, mrow, khalf);
    v2f b = load_bfrag(S1, NN, k0, n0, ncol, khalf);
    acc = wmma_f32_k4(a, b, acc);
  }
#pragma unroll 4
  for (int k0 = 0; k0 < DCH; k0 += 4) {        // source 2: A^2 x
    v2f a = load_afrag(Wg, 3 * DCH, m0, 2 * DCH + k0, mrow, khalf);
    v2f b = load_bfrag(S2, NN, k0, n0, ncol, khalf);
    acc = wmma_f32_k4(a, b, acc);
  }

  int n = n0 + ncol;
#pragma unroll
  for (int v = 0; v < 8; ++v) {                // new x -> LDS only
    int m = m0 + v + 8 * khalf;
    float gi = gg[m] * rsqrtf(gv[m] + EPSBN);
    float bi = ng[m] * rsqrtf(nv[m] + EPSBN);
    float h = acc[v] + bg[m];
    h = h * gi + (gb[m] - gm[m] * gi);
    h = h + S0[m * NN + n];
    lnx[m * LNXP + ncol] = h * bi + (nb[m] - nm[m] * bi);
  }

  __syncthreads();

#pragma unroll 4
  for (int k0 = 0; k0 < DCH; k0 += 4) {        // skip2 = Wsk2 @ newx (LDS B-frags)
    v2f a = load_afrag(Wsk2, DCH, m0, k0, mrow, khalf);
    const float* p = lnx + (k0 + 2 * khalf) * LNXP + ncol;
    v2f b; b.x = p[0]; b.y = p[LNXP];
    accs = wmma_f32_k4(a, b, accs);
  }

#pragma unroll
  for (int v = 0; v < 8; ++v) {
    int m = m0 + v + 8 * khalf;
    lsum[ncol * LSUMP + m] = accs[v] + bsk1[m] + bsk2[m];
  }

  __syncthreads();

  float* ob = out + (slice * NN + n0) * DCH;   // float4 RMW + output-BN affine
#pragma unroll
  for (int j = threadIdx.x * 4; j < 16 * DCH; j += 256 * 4) {
    int nn = j >> 7, dd = j & 127;
    float4 prev = *(float4*)(ob + j);
    const float* ls = lsum + nn * LSUMP + dd;
    float4 r;
    float inv = og[dd + 0] * rsqrtf(ov[dd + 0] + EPSBN);
    r.x = (prev.x + ls[0]) * inv + (obt[dd + 0] - om[dd + 0] * inv);
    inv = og[dd + 1] * rsqrtf(ov[dd + 1] + EPSBN);
    r.y = (prev.y + ls[1]) * inv + (obt[dd + 1] - om[dd + 1] * inv);
    inv = og[dd + 2] * rsqrtf(ov[dd + 2] + EPSBN);
    r.z = (prev.z + ls[2]) * inv + (obt[dd + 2] - om[dd + 2] * inv);
    inv = og[dd + 3] * rsqrtf(ov[dd + 3] + EPSBN);
    r.w = (prev.w + ls[3]) * inv + (obt[dd + 3] - om[dd + 3] * inv);
    *(float4*)(ob + j) = r;
  }
}

// ---------------------------------------------------------------------------
extern "C" void kernel_launch(void* const* d_in, const int* in_sizes, int n_in,
                              void* d_out, int out_size, void* d_ws, size_t ws_size,
                              hipStream_t stream) {
  const float* x    = (const float*)d_in[0];
  const float* adj  = (const float*)d_in[1];
  const float* Ws   = (const float*)d_in[2];
  const float* bs   = (const float*)d_in[3];
  const float* Wsk  = (const float*)d_in[4];   // [3,128,128]
  const float* bsk  = (const float*)d_in[5];   // [3,128]
  const float* Wgc  = (const float*)d_in[6];   // [2,128,384]
  const float* bgc  = (const float*)d_in[7];   // [2,128]
  const float* gg   = (const float*)d_in[8];
  const float* gb   = (const float*)d_in[9];
  const float* gm   = (const float*)d_in[10];
  const float* gv   = (const float*)d_in[11];
  const float* ng   = (const float*)d_in[12];
  const float* nb   = (const float*)d_in[13];
  const float* nm   = (const float*)d_in[14];
  const float* nv   = (const float*)d_in[15];
  const float* og   = (const float*)d_in[16];
  const float* obt  = (const float*)d_in[17];
  const float* om   = (const float*)d_in[18];
  const float* ov   = (const float*)d_in[19];

  float* out = (float*)d_out;
  float* P0 = (float*)d_ws;          // 3 ping-pong tensors, 201.3 MB each
  float* P1 = P0 + BUFN;
  float* P2 = P1 + BUFN;

  dim3 blk(256);
  const int GH = NBT * 8;            // 6144 blocks: (slice, 16-row strip)
  const int GG = NBT * 32;           // 24576 blocks: (slice, 16-col strip)

  // start conv + x1 + x2
  k_start_hop2<<<GH, blk, 0, stream>>>(x, Ws, bs, adj, P0, P1, P2);

  // gc step 0 (+skip0): new x -> P2, out initialized
  k_gc0<<<GG, blk, 0, stream>>>(P0, P1, P2, Wgc, bgc, Wsk, bsk,
                                gg, gb, gm, gv, ng, nb, nm, nv, out);

  // x1' -> P0, x2' -> P1 from new x = P2
  k_hop2<<<GH, blk, 0, stream>>>(P2, adj, P0, P1);

  // gc step 1 (+skip1+skip2+output BN): out finalized
  k_gcF<<<GG, blk, 0, stream>>>(P2, P0, P1,
                                Wgc + 128 * 384, bgc + 128,
                                Wsk + 128 * 128, bsk + 128,
                                Wsk + 2 * 128 * 128, bsk + 2 * 128,
                                gg + 128, gb + 128, gm + 128, gv + 128,
                                ng + 128, nb + 128, nm + 128, nv + 128,
                                og, obt, om, ov, out);
}